// Pixelwise_77919296684103
// MI455X (gfx1250) — compile-verified
//
#include <hip/hip_runtime.h>
#include <hip/hip_bf16.h>
#include <math.h>

// ---------------- constants (match reference) ----------------
#define NCODE  10000
#define KCH    3
#define NTILE  625           // NCODE / 16
#define CHUNKS 5             // code-dim split per decode block (one chunk per wave)
#define CT     (NTILE / CHUNKS)   // 125 tiles per chunk
static __device__ __host__ constexpr double D_DMAX   = 10000.0;
static __device__ __host__ constexpr double D_CLIGHT = 299792458.0 * 1000.0;   // mm/s
static __device__ __host__ constexpr double D_TAU    = (2.0 * D_DMAX) / D_CLIGHT;
static __device__ __host__ constexpr double D_DT     = D_TAU / (double)NCODE;
static __device__ __host__ constexpr double D_PAMB   = 1.0e6;
// GAMMA*MEAN_BETA*T_INT scale applied to BVals in fp32 by the reference
static __device__ __host__ constexpr double D_GAMMA  = 1.0 / (1.0e-4 * 0.1 * (1.0e6 + 1.0e9));
static __device__ __host__ constexpr double D_BSCALE = D_GAMMA * 1.0e-4 * 0.1;

typedef float v2f __attribute__((ext_vector_type(2)));
typedef float v8f __attribute__((ext_vector_type(8)));

// ---------------------------------------------------------------------------
// K1: CorrFsB[k][n] = DT * sum_m Mod[m,k]*Dem[(m+n)%N,k]   (f64 accumulate)
// Circular sum split into two linear segments (no per-iteration wrap),
// unrolled x4 with independent accumulators to break the f64 FMA chain.
// Demod column staged in LDS; Mod index is block-uniform only in the first segment.
// ---------------------------------------------------------------------------
__global__ __launch_bounds__(256) void corr_kernel(const float* __restrict__ Mod,
                                                   const float* __restrict__ Dem,
                                                   float* __restrict__ corr) {
    __shared__ float sDem[NCODE];
    const int k     = blockIdx.x / 40;
    const int nbase = (blockIdx.x % 40) * 256;
    for (int i = threadIdx.x; i < NCODE; i += 256)
        sDem[i] = Dem[i * KCH + k];
    __syncthreads();
    const int n = nbase + threadIdx.x;
    if (n >= NCODE) return;

    double a0 = 0.0, a1 = 0.0, a2 = 0.0, a3 = 0.0;
    auto seg = [&](int m0, int d0, int L) {
        int i = 0;
        for (; i + 4 <= L; i += 4) {
            a0 = fma((double)Mod[(m0 + i + 0) * KCH + k], (double)sDem[d0 + i + 0], a0);
            a1 = fma((double)Mod[(m0 + i + 1) * KCH + k], (double)sDem[d0 + i + 1], a1);
            a2 = fma((double)Mod[(m0 + i + 2) * KCH + k], (double)sDem[d0 + i + 2], a2);
            a3 = fma((double)Mod[(m0 + i + 3) * KCH + k], (double)sDem[d0 + i + 3], a3);
        }
        for (; i < L; ++i)
            a0 = fma((double)Mod[(m0 + i) * KCH + k], (double)sDem[d0 + i], a0);
    };
    seg(0, n, NCODE - n);        // m in [0, N-n): Dem index n+m
    seg(NCODE - n, 0, n);        // m in [N-n, N): Dem index m-(N-n)
    const double acc = (a0 + a1) + (a2 + a3);
    corr[k * NCODE + n] = (float)(acc * D_DT);
}

// ---------------------------------------------------------------------------
// K2: per-column mean / std (ddof=1) of CorrFsB, and ambient term
//     amb[k] = fp32( P_AMBIENT * fp32(fp32(sum Dem[:,k]) * fp32(DT)) )
// stats layout: [0..2]=mean, [3..5]=std, [6..8]=amb
// ---------------------------------------------------------------------------
__global__ __launch_bounds__(256) void stats_kernel(const float* __restrict__ corr,
                                                    const float* __restrict__ Dem,
                                                    float* __restrict__ stats) {
    const int k = blockIdx.x;
    const int t = threadIdx.x;
    __shared__ double sh0[256], sh1[256], sh2[256];
    double s = 0.0, ss = 0.0, kp = 0.0;
    for (int i = t; i < NCODE; i += 256) {
        double v = (double)corr[k * NCODE + i];
        s += v; ss += v * v;
        kp += (double)Dem[i * KCH + k];
    }
    sh0[t] = s; sh1[t] = ss; sh2[t] = kp;
    __syncthreads();
    for (int off = 128; off > 0; off >>= 1) {
        if (t < off) { sh0[t] += sh0[t+off]; sh1[t] += sh1[t+off]; sh2[t] += sh2[t+off]; }
        __syncthreads();
    }
    if (t == 0) {
        double mean = sh0[0] / (double)NCODE;
        double var  = (sh1[0] - (double)NCODE * mean * mean) / (double)(NCODE - 1);
        stats[k]     = (float)mean;
        stats[3 + k] = (float)sqrt(var);
        float kf = (float)sh2[0];            // fp32 sum like reference
        kf *= (float)D_DT;                   // * DT in fp32
        stats[6 + k] = (float)D_PAMB * kf;   // P_AMBIENT * kappa in fp32
    }
}

// ---------------------------------------------------------------------------
// K3: pack B in native WMMA 16x16x4 f32 B layout, one v2f per lane per tile:
//   Bpack[t*64 + lane*2 + v] = B[K = v + 2*(lane>=16), N = t*16 + (lane&15)]
//   rows K=0..2 are NormCorr channels, K=3 carries -0.5*||c_n||^2
// ---------------------------------------------------------------------------
__global__ __launch_bounds__(256) void packB_kernel(const float* __restrict__ corr,
                                                    const float* __restrict__ stats,
                                                    float* __restrict__ Bpack) {
    const int n = blockIdx.x * 256 + threadIdx.x;
    if (n >= NCODE) return;
    float nc0 = (corr[0 * NCODE + n] - stats[0]) / stats[3];
    float nc1 = (corr[1 * NCODE + n] - stats[1]) / stats[4];
    float nc2 = (corr[2 * NCODE + n] - stats[2]) / stats[5];
    float c2  = nc0 * nc0 + nc1 * nc1 + nc2 * nc2;
    const int t = n >> 4, l = n & 15;
    float* base = Bpack + t * 64;
    base[l * 2 + 0]        = nc0;          // K=0 (lanes 0-15, vgpr0)
    base[l * 2 + 1]        = nc1;          // K=1 (lanes 0-15, vgpr1)
    base[(16 + l) * 2 + 0] = nc2;          // K=2 (lanes 16-31, vgpr0)
    base[(16 + l) * 2 + 1] = -0.5f * c2;   // K=3 (lanes 16-31, vgpr1)
}

// ---------------------------------------------------------------------------
// K4: pack A (decode LUT rows r=0..9999) in WMMA A layout.
//   b_k = fp32(Corr[r,k] + amb_k) * BSCALE  (replicates reference fp32 path)
//   standardize over k (ddof=1), 4th K slot = 1.0 (picks up the -0.5*c2 row)
// ---------------------------------------------------------------------------
__global__ __launch_bounds__(256) void packA_kernel(const float* __restrict__ corr,
                                                    const float* __restrict__ stats,
                                                    float* __restrict__ Apack) {
    const int r = blockIdx.x * 256 + threadIdx.x;
    if (r >= NCODE) return;
    const float sf = (float)D_BSCALE;
    float b0 = (corr[0 * NCODE + r] + stats[6]) * sf;
    float b1 = (corr[1 * NCODE + r] + stats[7]) * sf;
    float b2 = (corr[2 * NCODE + r] + stats[8]) * sf;
    float mu = (b0 + b1 + b2) * (1.0f / 3.0f);
    float d0 = b0 - mu, d1 = b1 - mu, d2 = b2 - mu;
    float sd = sqrtf((d0 * d0 + d1 * d1 + d2 * d2) * 0.5f);   // ddof=1, n=3
    const int t = r >> 4, l = r & 15;
    float* base = Apack + t * 64;
    base[l * 2 + 0]        = d0 / sd;   // M=l, K=0
    base[l * 2 + 1]        = d1 / sd;   // M=l, K=1
    base[(16 + l) * 2 + 0] = d2 / sd;   // M=l, K=2
    base[(16 + l) * 2 + 1] = 1.0f;      // M=l, K=3  (bias multiplier)
}

// ---------------------------------------------------------------------------
// K5: WMMA decode. One block (5 waves) per row-tile of 16 LUT rows; each
// wave handles a 125-tile chunk of the code dimension (3125 waves total).
// D[M,N] = b_M . c_N - 0.5*||c_N||^2 ; per-lane running argmax over the 8
// D-rows it holds; B tile loads are software-pipelined so the pre-WMMA wait
// targets a load issued one iteration earlier. Cross-lane argmax via shfl,
// then the 5 chunk candidates merge through LDS (ascending chunk order
// preserves argmin first-index tie-breaking).
// ---------------------------------------------------------------------------
__global__ __launch_bounds__(160) void decode_kernel(const float* __restrict__ Apack,
                                                     const float* __restrict__ Bpack,
                                                     int* __restrict__ dec) {
    __shared__ float sVal[CHUNKS][16];
    __shared__ int   sIdx[CHUNKS][16];
    const int lane    = threadIdx.x & 31;
    const int wave    = threadIdx.x >> 5;      // chunk id 0..4
    const int rowTile = blockIdx.x;            // 0..624
    const v2f a = *(const v2f*)(Apack + rowTile * 64 + lane * 2);

    float bestV[8];
    int   bestI[8];
#pragma unroll
    for (int j = 0; j < 8; ++j) { bestV[j] = -__builtin_inff(); bestI[j] = 0; }

    const int nl = lane & 15;
    const int t0 = wave * CT;
    const float* bbase = Bpack + (size_t)t0 * 64 + lane * 2;

    v2f bcur = *(const v2f*)(bbase);
    for (int t = 0; t < CT; ++t) {
        const int tn = (t + 1 < CT) ? (t + 1) : t;          // uniform, branchless
        const v2f bnext = *(const v2f*)(bbase + (size_t)tn * 64);
        v8f cz = {0.f, 0.f, 0.f, 0.f, 0.f, 0.f, 0.f, 0.f};
        // D = A(16x4) x B(4x16) + 0   -> v_wmma_f32_16x16x4_f32
        v8f d = __builtin_amdgcn_wmma_f32_16x16x4_f32(
            false, a, false, bcur, (short)0, cz, false, false);
        const int ncand = ((t0 + t) << 4) + nl;
#pragma unroll
        for (int j = 0; j < 8; ++j)
            if (d[j] > bestV[j]) { bestV[j] = d[j]; bestI[j] = ncand; }
        bcur = bnext;
    }

    // reduce across the 16 lanes that share each row (max value, lowest index)
#pragma unroll
    for (int j = 0; j < 8; ++j) {
        float v = bestV[j];
        int   i = bestI[j];
        for (int m = 8; m >= 1; m >>= 1) {       // xor<16 stays within half-wave
            float ov = __shfl_xor(v, m);
            int   oi = __shfl_xor(i, m);
            if (ov > v || (ov == v && oi < i)) { v = ov; i = oi; }
        }
        bestV[j] = v; bestI[j] = i;
    }
    if ((lane & 15) == 0) {
        const int rbase = (lane >> 4) << 3;      // lane0 -> rows 0..7, lane16 -> 8..15
#pragma unroll
        for (int j = 0; j < 8; ++j) {
            sVal[wave][rbase + j] = bestV[j];
            sIdx[wave][rbase + j] = bestI[j];
        }
    }
    __syncthreads();
    if (threadIdx.x < 16) {
        const int row = threadIdx.x;
        float v = sVal[0][row];
        int   i = sIdx[0][row];
#pragma unroll
        for (int w = 1; w < CHUNKS; ++w)         // ascending n -> strict > keeps first
            if (sVal[w][row] > v) { v = sVal[w][row]; i = sIdx[w][row]; }
        dec[rowTile * 16 + row] = i;
    }
}

// ---------------------------------------------------------------------------
// K6: gather: out[p] = (float)dec[ clip(rint(depth[p]), 0, NCODE-1) ]
// rintf = round-half-even, matching jnp.round.
// ---------------------------------------------------------------------------
__global__ __launch_bounds__(256) void gather_kernel(const float* __restrict__ depth,
                                                     const int* __restrict__ dec,
                                                     float* __restrict__ out, int n) {
    const int p = blockIdx.x * 256 + threadIdx.x;
    if (p < n) {
        int idx = (int)rintf(depth[p]);
        idx = idx < 0 ? 0 : (idx > NCODE - 1 ? NCODE - 1 : idx);
        out[p] = (float)dec[idx];
    }
}

// ---------------------------------------------------------------------------
extern "C" void kernel_launch(void* const* d_in, const int* in_sizes, int n_in,
                              void* d_out, int out_size, void* d_ws, size_t ws_size,
                              hipStream_t stream) {
    const float* depth = (const float*)d_in[0];   // [1,120,160]
    const float* Mod   = (const float*)d_in[1];   // [10000,3] row-major
    const float* Dem   = (const float*)d_in[2];   // [10000,3] row-major
    float* out = (float*)d_out;

    char* ws = (char*)d_ws;
    float* corr  = (float*)(ws + 0);        // 30000 f  (120000 B)
    float* stats = (float*)(ws + 120000);   // 9 f      (pad to 64 B)
    float* Bpack = (float*)(ws + 120064);   // 625*64 f (160000 B)
    float* Apack = (float*)(ws + 280064);   // 625*64 f (160000 B)
    int*   dec   = (int*)  (ws + 440064);   // 10000 i  (40000 B)

    corr_kernel  <<<KCH * 40, 256, 0, stream>>>(Mod, Dem, corr);
    stats_kernel <<<KCH,      256, 0, stream>>>(corr, Dem, stats);
    packB_kernel <<<40,       256, 0, stream>>>(corr, stats, Bpack);
    packA_kernel <<<40,       256, 0, stream>>>(corr, stats, Apack);
    decode_kernel<<<NTILE,    32 * CHUNKS, 0, stream>>>(Apack, Bpack, dec);
    gather_kernel<<<(out_size + 255) / 256, 256, 0, stream>>>(depth, dec, out, out_size);
}